// ChittaEncoder_34299608826313
// MI455X (gfx1250) — compile-verified
//
#include <hip/hip_runtime.h>
#include <math.h>

// MI455X / gfx1250, wave32.
// Dominant op: sims = q @ seed^T -> streams 409.6 MB of seed_memory, HBM-bound
// (~17.6 us at 23.3 TB/s). We keep f32 precision (V_WMMA_F32_16X16X4_F32) and
// fix the data path: seed/q tiles are DMA'd into LDS by the Tensor Data Mover
// (TENSOR_LOAD_TO_LDS, TENSORcnt-tracked, double buffered) so all HBM traffic
// is linear, and WMMA fragments come from conflict-free ds_load_b64.

typedef __attribute__((ext_vector_type(2))) float v2f;
typedef __attribute__((ext_vector_type(4))) float v4f;
typedef __attribute__((ext_vector_type(8))) float v8f;

typedef __attribute__((ext_vector_type(4))) unsigned int u32x4;
typedef __attribute__((ext_vector_type(8))) int          i32x8;
typedef __attribute__((ext_vector_type(4))) int          i32x4;

#define KC 64          // K-chunk (floats) per TDM tile
#define KP 66          // padded LDS row stride (dwords): 16 frag rows -> distinct banks

// ---------------------------------------------------------------------------
// Issue one 2-D TDM tile load: rows x KC f32 elements, row stride `stride0`
// elements, into LDS at byte offset lds_addr with +2-dword padding per row
// (pad_interval=5 -> every 256B, pad_amount=1 -> 2 dwords). D# layout per
// CDNA5 ISA 8.3/8.4: group0 = {flags, lds_addr, global_addr, type=2},
// group1 = {ctrl, tensor_dim0/1, tile_dim0/1/2, dim0_stride}.
// This toolchain's builtin takes 6 args (g0, g1, g2, g3, g4, cpol).
// ---------------------------------------------------------------------------
__device__ __forceinline__ void tdm_load_2d(unsigned lds_addr, const float* gptr,
                                            unsigned tensor_d0, unsigned tensor_d1,
                                            unsigned tile_d0, unsigned tile_d1,
                                            unsigned stride0)
{
    unsigned long long ga = (unsigned long long)(const void*)gptr;
    u32x4 g0;
    g0[0] = 1u;                                             // count=1, user descriptor
    g0[1] = lds_addr;                                       // LDS byte address
    g0[2] = (unsigned)(ga & 0xFFFFFFFFu);                   // global_addr[31:0]
    g0[3] = (unsigned)((ga >> 32) & 0x01FFFFFFu)            // global_addr[56:32]
          | 0x80000000u;                                    // type=2 ("image")
    i32x8 g1;
    g1[0] = (int)0x03520000u;   // data_size=4B | pad_enable | pad_interval=5 | pad_amount=1
    g1[1] = (int)((tensor_d0 & 0xFFFFu) << 16);             // tensor_dim0[15:0]
    g1[2] = (int)((tensor_d0 >> 16) | ((tensor_d1 & 0xFFFFu) << 16));
    g1[3] = (int)((tensor_d1 >> 16) | (tile_d0 << 16));     // tile_dim0
    g1[4] = (int)(tile_d1 & 0xFFFFu);                       // tile_dim1 (tile_dim2=0)
    g1[5] = (int)stride0;                                   // tensor_dim0_stride[31:0]
    g1[6] = 0;
    g1[7] = 0;
    i32x4 z4 = {0, 0, 0, 0};                                // groups 2/3 unused (2-D)
    i32x8 z8 = {0, 0, 0, 0, 0, 0, 0, 0};
    __builtin_amdgcn_tensor_load_to_lds(g0, g1, z4, z4, z8, 0);
}

// ---------------------------------------------------------------------------
// sims kernel: block = 256 threads (8 waves) computes a [64 x 32] logits tile.
// Waves: tm = w&3 (M subtile), tnn = w>>2 (N subtile). K pipelined in KC=64
// chunks via TDM double buffering; 16 WMMAs per wave per chunk.
// Epilogue: logits = (acc * sigmoid((karma+0.3)*10) + vbias[type]) * 2
// ---------------------------------------------------------------------------
__global__ void sims_wmma_tdm(const float* __restrict__ q,
                              const float* __restrict__ seed,
                              const float* __restrict__ karma,
                              const int* __restrict__ vtypes,
                              const float* __restrict__ vbias,
                              float* __restrict__ out,
                              int N, int K)
{
    __shared__ float At[2][64 * KP];   // q tile:   64 rows x KC (padded)
    __shared__ float Bt[2][32 * KP];   // seed tile: <=32 rows x KC (padded)

    const int lane   = threadIdx.x & 31;
    const int waveId = threadIdx.x >> 5;
    const int half   = lane >> 4;
    const int l15    = lane & 15;

    const int tm  = waveId & 3;        // 0..3  -> M rows tm*16..tm*16+15
    const int tnn = waveId >> 2;       // 0..1  -> N cols tnn*16..tnn*16+15
    const int nb0 = blockIdx.x * 32;
    const unsigned brows = (unsigned)((N - nb0) < 32 ? (N - nb0) : 32);
    const int nchunks = K / KC;        // 2048/64 = 32

    // LDS byte offsets of the buffers (shared pointers are LDS offsets)
    const unsigned aoff0 = (unsigned)(unsigned long long)(&At[0][0]);
    const unsigned aoff1 = (unsigned)(unsigned long long)(&At[1][0]);
    const unsigned boff0 = (unsigned)(unsigned long long)(&Bt[0][0]);
    const unsigned boff1 = (unsigned)(unsigned long long)(&Bt[1][0]);

    // Prologue: wave 0 issues chunk 0 into buffer 0
    if (waveId == 0) {
        tdm_load_2d(aoff0, q, (unsigned)K, 64u, KC, 64u, (unsigned)K);
        tdm_load_2d(boff0, seed + (size_t)nb0 * K, (unsigned)K, (unsigned)N,
                    KC, brows, (unsigned)K);
    }

    v8f acc = {};
    for (int ic = 0; ic < nchunks; ++ic) {
        const int buf = ic & 1;
        if (waveId == 0) {
            if (ic + 1 < nchunks) {   // prefetch next chunk into the other buffer
                const int k1 = (ic + 1) * KC;
                tdm_load_2d(buf ? aoff0 : aoff1, q + k1,
                            (unsigned)K, 64u, KC, 64u, (unsigned)K);
                tdm_load_2d(buf ? boff0 : boff1, seed + (size_t)nb0 * K + k1,
                            (unsigned)K, (unsigned)N, KC, brows, (unsigned)K);
                __builtin_amdgcn_s_wait_tensorcnt(2);  // retire chunk ic (next 2 in flight)
            } else {
                __builtin_amdgcn_s_wait_tensorcnt(0);
            }
        }
        __syncthreads();               // chunk ic visible to all 8 waves

        const float* Arow = &At[buf][(tm * 16 + l15) * KP];
        const float* Brow = &Bt[buf][(tnn * 16 + l15) * KP];
#pragma unroll
        for (int kk = 0; kk < KC; kk += 4) {
            const int kb = kk + (half << 1);   // lanes 0-15: K=kk,kk+1 ; 16-31: kk+2,kk+3
            v2f a = *(const v2f*)(Arow + kb);  // ds_load_b64, conflict-free (stride 66)
            v2f b = *(const v2f*)(Brow + kb);
            acc = __builtin_amdgcn_wmma_f32_16x16x4_f32(false, a, false, b,
                                                        (short)0, acc, false, false);
        }
        __syncthreads();               // done reading buf before it is refilled
    }

    const int col = nb0 + tnn * 16 + l15;
    if (col < N) {
        const float g  = 1.0f / (1.0f + expf(-(karma[col] + 0.3f) * 10.0f));
        const float vb = vbias[vtypes[col]];
#pragma unroll
        for (int r = 0; r < 8; ++r) {
            const int m = tm * 16 + r + (half << 3);
            out[(size_t)m * N + col] = (acc[r] * g + vb) * 2.0f;  // 1/T = 2 folded in
        }
    }
}

// ---------------------------------------------------------------------------
// MLP GEMM (weights are [K,N] row-major, loads coalesced): one 16x16 tile per
// wave. EPI: 0 = +bias, 1 = +bias then exact erf GELU.
// ---------------------------------------------------------------------------
template <int EPI>
__global__ void wmma_gemm_f32(const float* __restrict__ A,
                              const float* __restrict__ B,
                              const float* __restrict__ bias,
                              float* __restrict__ C,
                              int M, int N, int K)
{
    const int lane   = threadIdx.x & 31;
    const int waveId = threadIdx.x >> 5;
    const int tilesN = N >> 4;
    const int tile   = blockIdx.x * (blockDim.x >> 5) + waveId;
    if (tile >= (M >> 4) * tilesN) return;

    const int tm = tile / tilesN;
    const int tn = tile - tm * tilesN;
    const int m0 = tm << 4, n0 = tn << 4;

    const int half = lane >> 4;
    const int l15  = lane & 15;
    const float* Arow = A + (size_t)(m0 + l15) * K;
    const float* Bcol = B + (n0 + l15);
    const int bcol = n0 + l15;

    v8f acc = {};
#pragma unroll 8
    for (int k = 0; k < K; k += 4) {
        const int kb = k + (half << 1);
        v2f a = *(const v2f*)(Arow + kb);
        v2f b;
        b.x = Bcol[(size_t)kb * N];
        b.y = Bcol[(size_t)(kb + 1) * N];
        acc = __builtin_amdgcn_wmma_f32_16x16x4_f32(false, a, false, b,
                                                    (short)0, acc, false, false);
    }

    const float bb = bias[bcol];
#pragma unroll
    for (int r = 0; r < 8; ++r) {
        const int m = m0 + r + (half << 3);
        float v = acc[r] + bb;
        if (EPI == 1)   // exact (erf-based) GELU, matches approximate=False
            v = 0.5f * v * (1.0f + erff(v * 0.70710678118654752f));
        C[(size_t)m * N + bcol] = v;
    }
}

// ---------------------------------------------------------------------------
// In-place numerically-stable softmax over each row of [B, N]
// ---------------------------------------------------------------------------
__global__ void softmax_rows(float* __restrict__ attn, int N)
{
    __shared__ float red[256];
    float* row = attn + (size_t)blockIdx.x * N;
    const int t = threadIdx.x;

    float m = -INFINITY;
    for (int i = t; i < N; i += 256) m = fmaxf(m, row[i]);
    red[t] = m; __syncthreads();
    for (int s = 128; s > 0; s >>= 1) {
        if (t < s) red[t] = fmaxf(red[t], red[t + s]);
        __syncthreads();
    }
    m = red[0]; __syncthreads();

    float sum = 0.0f;
    for (int i = t; i < N; i += 256) {
        float e = expf(row[i] - m);
        row[i] = e;
        sum += e;
    }
    red[t] = sum; __syncthreads();
    for (int s = 128; s > 0; s >>= 1) {
        if (t < s) red[t] += red[t + s];
        __syncthreads();
    }
    const float inv = 1.0f / red[0];
    __syncthreads();
    for (int i = t; i < N; i += 256) row[i] *= inv;
}

// ---------------------------------------------------------------------------
// Deterministic top-K per row: K sequential arg-max passes with strict
// (value desc, index asc) ordering -> matches jax.lax.top_k semantics.
// ---------------------------------------------------------------------------
__global__ void topk_rows(const float* __restrict__ attn, int N, int TOPK,
                          float* __restrict__ out_idx_f, int* __restrict__ out_idx_i)
{
    __shared__ float sv[256];
    __shared__ int   si[256];
    __shared__ float prevV;
    __shared__ int   prevI;

    const int t = threadIdx.x;
    const float* row = attn + (size_t)blockIdx.x * N;
    if (t == 0) { prevV = INFINITY; prevI = -1; }
    __syncthreads();

    for (int j = 0; j < TOPK; ++j) {
        const float pv = prevV;
        const int   pi = prevI;
        float bv = -INFINITY;
        int   bi = N;
        for (int i = t; i < N; i += 256) {
            const float v = row[i];
            const bool below = (v < pv) || (v == pv && i > pi);
            if (below && (v > bv || (v == bv && i < bi))) { bv = v; bi = i; }
        }
        sv[t] = bv; si[t] = bi;
        __syncthreads();
        for (int s = 128; s > 0; s >>= 1) {
            if (t < s) {
                if (sv[t + s] > sv[t] || (sv[t + s] == sv[t] && si[t + s] < si[t])) {
                    sv[t] = sv[t + s]; si[t] = si[t + s];
                }
            }
            __syncthreads();
        }
        if (t == 0) {
            prevV = sv[0]; prevI = si[0];
            out_idx_f[(size_t)blockIdx.x * TOPK + j] = (float)si[0];
            out_idx_i[(size_t)blockIdx.x * TOPK + j] = si[0];
        }
        __syncthreads();
    }
}

// ---------------------------------------------------------------------------
// chitta_field[b,k,:] = seed_memory[idx[b,k], :]   (float4 copies)
// ---------------------------------------------------------------------------
__global__ void gather_seeds(const float* __restrict__ seed,
                             const int* __restrict__ idx,
                             float* __restrict__ out, int D, int TOPK)
{
    const int b = blockIdx.y, k = blockIdx.x;
    const int s = idx[b * TOPK + k];
    const v4f* src = (const v4f*)(seed + (size_t)s * D);
    v4f* dst = (v4f*)(out + ((size_t)b * TOPK + k) * D);
    const int n4 = D >> 2;
    for (int i = threadIdx.x; i < n4; i += blockDim.x) dst[i] = src[i];
}

// ---------------------------------------------------------------------------
extern "C" void kernel_launch(void* const* d_in, const int* in_sizes, int n_in,
                              void* d_out, int out_size, void* d_ws, size_t ws_size,
                              hipStream_t stream)
{
    (void)in_sizes; (void)n_in; (void)out_size; (void)ws_size;

    const float* query  = (const float*)d_in[0];  // [64, 2048]
    const int*   vtypes = (const int*)  d_in[1];  // [50000]
    const float* seed   = (const float*)d_in[2];  // [50000, 2048]
    const float* karma  = (const float*)d_in[3];  // [50000]
    const float* vbias  = (const float*)d_in[4];  // [5]
    const float* W1     = (const float*)d_in[5];  // [2048, 4096]
    const float* b1     = (const float*)d_in[6];  // [4096]
    const float* W2     = (const float*)d_in[7];  // [4096, 2048]
    const float* b2     = (const float*)d_in[8];  // [2048]

    const int B = 64, N = 50000, D = 2048, H = 4096, TOPK = 50;

    // workspace: h [64x4096], q [64x2048], int top-k indices [64x50]
    float* h     = (float*)d_ws;
    float* q     = h + (size_t)B * H;
    int*   idx_i = (int*)(q + (size_t)B * D);

    // outputs concatenated flat: chitta_field, attention_weights, topk_indices
    float* chitta = (float*)d_out;                  // [64, 50, 2048]
    float* attn   = chitta + (size_t)B * TOPK * D;  // [64, 50000]
    float* idx_f  = attn + (size_t)B * N;           // [64, 50]

    // h = gelu(query @ W1 + b1): 4*256 = 1024 tiles / 8 waves -> 128 blocks
    wmma_gemm_f32<1><<<128, 256, 0, stream>>>(query, W1, b1, h, B, H, D);
    // q = h @ W2 + b2: 4*128 = 512 tiles -> 64 blocks
    wmma_gemm_f32<0><<<64, 256, 0, stream>>>(h, W2, b2, q, B, D, H);
    // logits via TDM-pipelined WMMA: ceil(50000/32) = 1563 blocks
    sims_wmma_tdm<<<1563, 256, 0, stream>>>(q, seed, karma, vtypes, vbias, attn, N, D);
    // softmax rows in-place -> attention_weights
    softmax_rows<<<B, 256, 0, stream>>>(attn, N);
    // top-50 (sorted desc, index-asc tiebreak)
    topk_rows<<<B, 256, 0, stream>>>(attn, N, TOPK, idx_f, idx_i);
    // gather retrieved seed embeddings
    gather_seeds<<<dim3(TOPK, B), 256, 0, stream>>>(seed, idx_i, chitta, D, TOPK);
}